// GeometricReconstructionLoss_4071628996825
// MI455X (gfx1250) — compile-verified
//
#include <hip/hip_runtime.h>
#include <hip/hip_bf16.h>

typedef __attribute__((ext_vector_type(2))) float v2f;
typedef __attribute__((ext_vector_type(8))) float v8f;

#define N_PTS   1024
#define NGROUPS 128     // B*G = 8*16
#define TPB     256
#define NWAVES  8
#define RT_PER_WAVE 8   // 64 row tiles / 8 waves
#define NCT     64      // 1024 / 16 column tiles

// One workgroup per (b,g) group. Computes cham_x + cham_y for the group
// via V_WMMA_F32_16X16X4_F32 with the distance formula folded into the MMA:
//   A = [x0,x1,x2,|x|^2], B = [-2y0,-2y1,-2y2,1], C = broadcast(|y|^2)
//   D = |x|^2 + |y|^2 - 2 x.y   (the squared pairwise distance tile)
__global__ __launch_bounds__(TPB)
void chamfer_group_kernel(const float* __restrict__ X,
                          const float* __restrict__ Y,
                          float* __restrict__ ws)
{
    __shared__ float ypad[N_PTS * 4];          // 16 KB: [-2y0,-2y1,-2y2,1]
    __shared__ float y2s[N_PTS];               //  4 KB: |y|^2
    __shared__ float colpart[NWAVES * N_PTS];  // 32 KB: per-wave column mins
    __shared__ float rowsum[64];               // per row-tile sum of row mins
    __shared__ float red[TPB];                 // final reduction scratch

    const int g    = blockIdx.x;
    const int tid  = threadIdx.x;
    const int wave = tid >> 5;
    const int lane = tid & 31;
    const int half = lane >> 4;   // 0: lanes 0-15, 1: lanes 16-31
    const int lp   = lane & 15;
    const float INF = __builtin_inff();

    const float* xg = X + (size_t)g * N_PTS * 3;
    const float* yg = Y + (size_t)g * N_PTS * 3;

    // ---- Stage Y into LDS, pre-scaled, plus squared norms ----
    for (int p = tid; p < N_PTS; p += TPB) {
        float y0 = yg[p * 3 + 0];
        float y1 = yg[p * 3 + 1];
        float y2v = yg[p * 3 + 2];
        ypad[p * 4 + 0] = -2.0f * y0;
        ypad[p * 4 + 1] = -2.0f * y1;
        ypad[p * 4 + 2] = -2.0f * y2v;
        ypad[p * 4 + 3] = 1.0f;
        y2s[p] = y0 * y0 + y1 * y1 + y2v * y2v;
    }
    __syncthreads();

    // ---- Load this wave's 8 A-tiles into registers ----
    // 16x4 f32 A layout: lanes 0-15 hold K0(K1), lanes 16-31 hold K2(K3).
    v2f A[RT_PER_WAVE];
#pragma unroll
    for (int r = 0; r < RT_PER_WAVE; ++r) {
        int row = (wave * RT_PER_WAVE + r) * 16 + lp;
        float x0 = xg[row * 3 + 0];
        float x1 = xg[row * 3 + 1];
        float x2 = xg[row * 3 + 2];
        float n2 = x0 * x0 + x1 * x1 + x2 * x2;
        v2f a;
        a.x = half ? x2 : x0;   // K0 or K2
        a.y = half ? n2 : x1;   // K1 or K3
        A[r] = a;
    }

    v8f rowmin[RT_PER_WAVE];
#pragma unroll
    for (int r = 0; r < RT_PER_WAVE; ++r) {
        v8f v = {INF, INF, INF, INF, INF, INF, INF, INF};
        rowmin[r] = v;
    }

    // ---- Sweep all 64 column tiles ----
    for (int ct = 0; ct < NCT; ++ct) {
        // B 4x16 layout mirrors A: lanes 0-15 = rows K0/K1, lanes 16-31 = K2/K3
        const v2f bv = *reinterpret_cast<const v2f*>(
            &ypad[(ct * 16 + lp) * 4 + half * 2]);
        const float yy = y2s[ct * 16 + lp];
        v8f cc = {yy, yy, yy, yy, yy, yy, yy, yy};

        float colp = INF;
#pragma unroll
        for (int r = 0; r < RT_PER_WAVE; ++r) {
            v8f d = __builtin_amdgcn_wmma_f32_16x16x4_f32(
                /*neg_a=*/false, A[r], /*neg_b=*/false, bv,
                /*c_mod=*/(short)0, cc, /*reuse_a=*/false, /*reuse_b=*/false);
#pragma unroll
            for (int i = 0; i < 8; ++i) {
                rowmin[r][i] = fminf(rowmin[r][i], d[i]);
                colp = fminf(colp, d[i]);
            }
        }
        // combine the two row-halves of this column (lane l vs l+16)
        colp = fminf(colp, __shfl_xor(colp, 16, 32));
        if (half == 0)
            colpart[wave * N_PTS + ct * 16 + lp] = colp;
    }

    // ---- Row-min cross-lane reduction (within 16-lane halves), sum per tile ----
#pragma unroll
    for (int r = 0; r < RT_PER_WAVE; ++r) {
        v8f rm = rowmin[r];
#pragma unroll
        for (int i = 0; i < 8; ++i) {
            float v = rm[i];
            v = fminf(v, __shfl_xor(v, 1, 32));
            v = fminf(v, __shfl_xor(v, 2, 32));
            v = fminf(v, __shfl_xor(v, 4, 32));
            v = fminf(v, __shfl_xor(v, 8, 32));
            rm[i] = v;
        }
        float s = ((rm[0] + rm[1]) + (rm[2] + rm[3])) +
                  ((rm[4] + rm[5]) + (rm[6] + rm[7]));
        s += __shfl_xor(s, 16, 32);          // rows 0-7 half + rows 8-15 half
        if (lane == 0)
            rowsum[wave * RT_PER_WAVE + r] = s;
    }
    __syncthreads();

    // ---- Deterministic final reduction for the group ----
    float part = 0.0f;
    for (int c = tid; c < N_PTS; c += TPB) {
        float m = colpart[c];
#pragma unroll
        for (int w = 1; w < NWAVES; ++w)
            m = fminf(m, colpart[w * N_PTS + c]);
        part += m;                            // cham_y contribution
    }
    if (tid < 64)
        part += rowsum[tid];                  // cham_x contribution
    red[tid] = part;
    __syncthreads();
    for (int s = TPB / 2; s > 0; s >>= 1) {
        if (tid < s) red[tid] += red[tid + s];
        __syncthreads();
    }
    if (tid == 0)
        ws[g] = red[0];
}

// Sum the 128 per-group results, apply batch mean (1/B = 0.125).
__global__ __launch_bounds__(NGROUPS)
void chamfer_finalize_kernel(const float* __restrict__ ws, float* __restrict__ out)
{
    __shared__ float r[NGROUPS];
    const int tid = threadIdx.x;
    r[tid] = ws[tid];
    __syncthreads();
    for (int s = NGROUPS / 2; s > 0; s >>= 1) {
        if (tid < s) r[tid] += r[tid + s];
        __syncthreads();
    }
    if (tid == 0)
        out[0] = r[0] * 0.125f;
}

extern "C" void kernel_launch(void* const* d_in, const int* in_sizes, int n_in,
                              void* d_out, int out_size, void* d_ws, size_t ws_size,
                              hipStream_t stream)
{
    const float* X = (const float*)d_in[0];        // [8,16,1024,3] f32
    const float* Y = (const float*)d_in[1];        // [8,16,1024,3] f32
    float* out = (float*)d_out;                    // scalar f32 loss
    float* ws  = (float*)d_ws;                     // >= 128 floats

    chamfer_group_kernel<<<NGROUPS, TPB, 0, stream>>>(X, Y, ws);
    chamfer_finalize_kernel<<<1, NGROUPS, 0, stream>>>(ws, out);
}